// PatchTransferCluster_2241972928976
// MI455X (gfx1250) — compile-verified
//
#include <hip/hip_runtime.h>

// ---------------------------------------------------------------------------
// Contextual-attention patch transfer for MI455X (gfx1250, wave32).
// fp32 WMMA (V_WMMA_F32_16X16X4_F32) for both GEMMs; double-buffered
// GLOBAL_LOAD_ASYNC_TO_LDS staging overlapped with WMMA via partial
// s_wait_asynccnt (async loads retire in order).
// ---------------------------------------------------------------------------

typedef float v2f __attribute__((ext_vector_type(2)));
typedef float v8f __attribute__((ext_vector_type(8)));

#define LTOT 4096          // 64*64 patch locations
#define CCH  128           // channels
#define EPS_TERM (1152.0f * 1.0e-4f)   // P*EPS, P = C*3*3
#define SCALE_F 10.0f

// ---- CDNA5 async memory->LDS helpers (ASYNCcnt tracked) -------------------
__device__ __forceinline__ unsigned lds_off(const void* p) {
  // gfx1250 aperture rule: LDS flat address low 32 bits == LDS byte offset
  return (unsigned)(unsigned long long)p;
}
__device__ __forceinline__ void async_ld_b32(unsigned lds, const void* g) {
  asm volatile("global_load_async_to_lds_b32 %0, %1, off"
               :: "v"(lds), "v"(g) : "memory");
}
__device__ __forceinline__ void async_ld_b128(unsigned lds, const void* g) {
  asm volatile("global_load_async_to_lds_b128 %0, %1, off"
               :: "v"(lds), "v"(g) : "memory");
}
template <int N>
__device__ __forceinline__ void wait_async() {
  asm volatile("s_wait_asynccnt %0" :: "i"(N) : "memory");
}

// ---------------- padding kernels ----------------
// f_pad: [C][66][66], 1-zero border around f downsampled by 2
__global__ void k_fpad(const float* __restrict__ f, float* __restrict__ fpad) {
  int idx = blockIdx.x * 256 + threadIdx.x;
  if (idx >= CCH * 66 * 66) return;
  int c = idx / 4356; int r = idx - c * 4356;
  int u = r / 66, v = r - u * 66;
  float val = 0.f;
  if (u >= 1 && u <= 64 && v >= 1 && v <= 64)
    val = f[c * 16384 + (2 * (u - 1)) * 128 + 2 * (v - 1)];
  fpad[idx] = val;
}

// b_pad: [C][130][130], 1-zero border around b (full res)
__global__ void k_bpad(const float* __restrict__ b, float* __restrict__ bpad) {
  int idx = blockIdx.x * 256 + threadIdx.x;
  if (idx >= CCH * 130 * 130) return;
  int c = idx / 16900; int r = idx - c * 16900;
  int u = r / 130, v = r - u * 130;
  float val = 0.f;
  if (u >= 1 && u <= 128 && v >= 1 && v <= 128)
    val = b[c * 16384 + (u - 1) * 128 + (v - 1)];
  bpad[idx] = val;
}

// ---------------- score GEMM: S[l,s] = sum_d sum_c fpad(l+d)*fpad(s+d) ------
// block tile 128(l) x 128(s); 8 waves (4x2), each 32x64 (2x4 WMMA accums);
// 72 K-chunks (9 offsets x 8), double-buffered async staging.
__global__ __launch_bounds__(256) void k_gemmS(const float* __restrict__ fpad,
                                               float* __restrict__ S) {
  __shared__ float AsT[2][16][132];  // k(channel-chunk) x l
  __shared__ float BsT[2][16][132];  // k(channel-chunk) x s
  const int tid  = threadIdx.x;
  const int lane = tid & 31, wid = tid >> 5;
  const int wm = wid & 3, wn = wid >> 2;       // m0 = 32*wm, n0 = 64*wn
  const int half = lane >> 4, lr = lane & 15;
  const int bs0 = blockIdx.x * 128;
  const int bl0 = blockIdx.y * 128;

  const v8f z8 = {0,0,0,0,0,0,0,0};
  v8f acc[2][4] = {{z8,z8,z8,z8},{z8,z8,z8,z8}};

  // staging map: 16 k-rows x 16 groups-of-4 (x2) -> per-lane contiguous B128
  const int cA = tid & 15;       // k row within chunk
  const int gA = tid >> 4;       // group id 0..15

  int lbase[2], sbase[2];
#pragma unroll
  for (int q = 0; q < 2; ++q) {
    int l = bl0 + (gA + 16 * q) * 4;          // aligned 4-groups never cross row
    lbase[q] = (l >> 6) * 66 + (l & 63);
    int s = bs0 + (gA + 16 * q) * 4;
    sbase[q] = (s >> 6) * 66 + (s & 63);
  }

  // issue one K-chunk's async loads into buffer (t&1); 4 wave-level async ops
  auto issue = [&](int t) {
    const int dd = t >> 3;               // 0..8 : (dy,dx) offset
    const int kk = (t & 7) << 4;         // 0..112: channel chunk
    const int dy = dd / 3, dx = dd - dy * 3;
    const float* fc = fpad + (kk + cA) * 4356 + dy * 66 + dx;
    float (*Ab)[132] = AsT[t & 1];
    float (*Bb)[132] = BsT[t & 1];
#pragma unroll
    for (int q = 0; q < 2; ++q) {
      async_ld_b128(lds_off(&Ab[cA][(gA + 16 * q) * 4]), fc + lbase[q]);
      async_ld_b128(lds_off(&Bb[cA][(gA + 16 * q) * 4]), fc + sbase[q]);
    }
  };

  issue(0);
  for (int t = 0; t < 72; ++t) {
    if (t + 1 < 72) { issue(t + 1); wait_async<4>(); }
    else            { wait_async<0>(); }
    __syncthreads();
    const float (*Ab)[132] = AsT[t & 1];
    const float (*Bb)[132] = BsT[t & 1];
#pragma unroll
    for (int k4 = 0; k4 < 16; k4 += 4) {
      const int kb = k4 + 2 * half;
      v2f a[2], b[4];
#pragma unroll
      for (int mi = 0; mi < 2; ++mi) {
        a[mi][0] = Ab[kb][wm * 32 + mi * 16 + lr];
        a[mi][1] = Ab[kb + 1][wm * 32 + mi * 16 + lr];
      }
#pragma unroll
      for (int ni = 0; ni < 4; ++ni) {
        b[ni][0] = Bb[kb][wn * 64 + ni * 16 + lr];
        b[ni][1] = Bb[kb + 1][wn * 64 + ni * 16 + lr];
      }
#pragma unroll
      for (int mi = 0; mi < 2; ++mi)
#pragma unroll
        for (int ni = 0; ni < 4; ++ni)
          acc[mi][ni] = __builtin_amdgcn_wmma_f32_16x16x4_f32(
              false, a[mi], false, b[ni], (short)0, acc[mi][ni], false, false);
    }
    __syncthreads();
  }
#pragma unroll
  for (int mi = 0; mi < 2; ++mi)
#pragma unroll
    for (int ni = 0; ni < 4; ++ni)
#pragma unroll
      for (int r = 0; r < 8; ++r) {
        int lm = bl0 + wm * 32 + mi * 16 + r + 8 * half;
        int sn = bs0 + wn * 64 + ni * 16 + lr;
        S[lm * LTOT + sn] = acc[mi][ni][r];
      }
}

// ---------------- stats: inv-norm from diag(S); mask flag mm[l] ------------
__global__ void k_stats(const float* __restrict__ S, const float* __restrict__ mask,
                        float* __restrict__ invn, float* __restrict__ mmv) {
  int l = blockIdx.x * 256 + threadIdx.x;
  if (l >= LTOT) return;
  invn[l] = rsqrtf(S[l * (LTOT + 1)] + EPS_TERM);
  int ly = l >> 6, lx = l & 63;
  float msum = 0.f;
  for (int dy = 0; dy < 3; ++dy) {
    int u = ly - 1 + dy; if (u < 0 || u > 63) continue;
    for (int dx = 0; dx < 3; ++dx) {
      int v = lx - 1 + dx; if (v < 0 || v > 63) continue;
      msum += mask[(2 * u) * 128 + 2 * v];
    }
  }
  mmv[l] = (msum == 0.f) ? 1.f : 0.f;
}

// ---------------- in-place masked softmax over l for each s ----------------
// S is symmetric, so row s of S == column s; after this S holds attn^T[s][l].
__global__ __launch_bounds__(256) void k_softmax(float* __restrict__ S,
                                                 const float* __restrict__ invn,
                                                 const float* __restrict__ mmv) {
  __shared__ float red[256];
  const int tid = threadIdx.x;
  float* row = S + (size_t)blockIdx.x * LTOT;
  float sc[16];
  float mmax = -1e30f;
#pragma unroll
  for (int t = 0; t < 16; ++t) {
    int l = tid + 256 * t;
    float x = row[l] * invn[l] * mmv[l] * SCALE_F;
    sc[t] = x;
    mmax = fmaxf(mmax, x);
  }
  red[tid] = mmax; __syncthreads();
  for (int off = 128; off > 0; off >>= 1) {
    if (tid < off) red[tid] = fmaxf(red[tid], red[tid + off]);
    __syncthreads();
  }
  float M = red[0]; __syncthreads();
  float lsum = 0.f;
#pragma unroll
  for (int t = 0; t < 16; ++t) { sc[t] = __expf(sc[t] - M); lsum += sc[t]; }
  red[tid] = lsum; __syncthreads();
  for (int off = 128; off > 0; off >>= 1) {
    if (tid < off) red[tid] += red[tid + off];
    __syncthreads();
  }
  float inv = 1.0f / red[0];
#pragma unroll
  for (int t = 0; t < 16; ++t) {
    int l = tid + 256 * t;
    row[l] = sc[t] * inv * mmv[l];
  }
}

// ---------------- recon GEMM: R_ij[c,s] = sum_l bpad[c,2ly+i,2lx+j]*attn[l,s]
// block tile 128(c) x 128(s), K = 4096 (l), 256 chunks, double-buffered async.
__global__ __launch_bounds__(256) void k_gemmR(const float* __restrict__ bpad,
                                               const float* __restrict__ attnT,
                                               float* __restrict__ R) {
  __shared__ float AsT[2][16][132];  // k(l-chunk) x c
  __shared__ float Bs2[2][128][17];  // s x k(l-chunk)   (k contiguous)
  const int tid  = threadIdx.x;
  const int lane = tid & 31, wid = tid >> 5;
  const int wm = wid & 3, wn = wid >> 2;
  const int half = lane >> 4, lr = lane & 15;
  const int bs0 = blockIdx.x * 128;
  const int ij  = blockIdx.y;
  const int di = ij >> 2, dj = ij & 3;

  const v8f z8 = {0,0,0,0,0,0,0,0};
  v8f acc[2][4] = {{z8,z8,z8,z8},{z8,z8,z8,z8}};

  const int lkA = tid & 15, c0 = tid >> 4;   // A: scalar gather (stride-2 x)
  const int lgB = tid & 3,  sr = tid >> 2;   // B: per-lane B128 along l

  // issue one K-chunk; 10 wave-level async ops (8x B32 gather + 2x B128)
  auto issue = [&](int t) {
    const int kk = t << 4;
    int l = kk + lkA;
    int ly = l >> 6, lx = l & 63;
    int badd = (2 * ly + di) * 130 + (2 * lx + dj);
    float (*Ab)[132] = AsT[t & 1];
    float (*Bb)[17]  = Bs2[t & 1];
#pragma unroll
    for (int jj = 0; jj < 8; ++jj) {
      int c = c0 + 16 * jj;
      async_ld_b32(lds_off(&Ab[lkA][c]), bpad + c * 16900 + badd);
    }
#pragma unroll
    for (int q = 0; q < 2; ++q) {
      int sI = sr + 64 * q;
      async_ld_b128(lds_off(&Bb[sI][lgB * 4]),
                    attnT + (size_t)(bs0 + sI) * LTOT + kk + lgB * 4);
    }
  };

  issue(0);
  for (int t = 0; t < 256; ++t) {
    if (t + 1 < 256) { issue(t + 1); wait_async<10>(); }
    else             { wait_async<0>(); }
    __syncthreads();
    const float (*Ab)[132] = AsT[t & 1];
    const float (*Bb)[17]  = Bs2[t & 1];
#pragma unroll
    for (int k4 = 0; k4 < 16; k4 += 4) {
      const int kb = k4 + 2 * half;
      v2f a[2], b[4];
#pragma unroll
      for (int mi = 0; mi < 2; ++mi) {
        a[mi][0] = Ab[kb][wm * 32 + mi * 16 + lr];
        a[mi][1] = Ab[kb + 1][wm * 32 + mi * 16 + lr];
      }
#pragma unroll
      for (int ni = 0; ni < 4; ++ni) {
        b[ni][0] = Bb[wn * 64 + ni * 16 + lr][kb];
        b[ni][1] = Bb[wn * 64 + ni * 16 + lr][kb + 1];
      }
#pragma unroll
      for (int mi = 0; mi < 2; ++mi)
#pragma unroll
        for (int ni = 0; ni < 4; ++ni)
          acc[mi][ni] = __builtin_amdgcn_wmma_f32_16x16x4_f32(
              false, a[mi], false, b[ni], (short)0, acc[mi][ni], false, false);
    }
    __syncthreads();
  }
  float* Rp = R + (size_t)ij * (CCH * LTOT);
#pragma unroll
  for (int mi = 0; mi < 2; ++mi)
#pragma unroll
    for (int ni = 0; ni < 4; ++ni)
#pragma unroll
      for (int r = 0; r < 8; ++r) {
        int cm = wm * 32 + mi * 16 + r + 8 * half;
        int sn = bs0 + wn * 64 + ni * 16 + lr;
        Rp[cm * LTOT + sn] = acc[mi][ni][r];
      }
}

// ---------------- combine: gather <=4 parity-valid (i,j) taps --------------
__global__ void k_combine(const float* __restrict__ R, float* __restrict__ out) {
  int idx = blockIdx.x * 256 + threadIdx.x;
  if (idx >= CCH * 16384) return;
  int c = idx >> 14;
  int rem = idx & 16383;
  int y = rem >> 7, x = rem & 127;
  float acc = 0.f;
  int i0 = (y + 1) & 1;
  int j0 = (x + 1) & 1;
  for (int i = i0; i < 4; i += 2) {
    int ty = y + 1 - i; if (ty < 0) break;
    int hd = ty >> 1;   if (hd > 63) continue;
    for (int j = j0; j < 4; j += 2) {
      int tx = x + 1 - j; if (tx < 0) break;
      int wd = tx >> 1;   if (wd > 63) continue;
      acc += R[(size_t)(i * 4 + j) * (CCH * LTOT) + c * LTOT + hd * 64 + wd];
    }
  }
  out[idx] = acc;
}

// ---------------------------------------------------------------------------
extern "C" void kernel_launch(void* const* d_in, const int* in_sizes, int n_in,
                              void* d_out, int out_size, void* d_ws, size_t ws_size,
                              hipStream_t stream) {
  const float* f    = (const float*)d_in[0];
  const float* b    = (const float*)d_in[1];
  const float* mask = (const float*)d_in[2];
  float* out = (float*)d_out;

  float* ws   = (float*)d_ws;
  float* fpad = ws;                         // 128*66*66      =   557,568 f
  float* bpad = fpad + 128 * 4356;          // 128*130*130    = 2,163,200 f
  float* S    = bpad + 128 * 16900;         // 4096*4096      = 16,777,216 f
  float* invn = S + (size_t)LTOT * LTOT;    // 4096
  float* mmv  = invn + LTOT;                // 4096
  float* R    = mmv + LTOT;                 // 16*128*4096    = 8,388,608 f
  // total ~111.6 MB of workspace, reused across batches (stream-ordered)

  for (int bi = 0; bi < 4; ++bi) {
    const float* fb = f + (size_t)bi * CCH * 16384;
    const float* bb = b + (size_t)bi * CCH * 16384;
    const float* mb = mask + (size_t)bi * 16384;
    float* ob = out + (size_t)bi * CCH * 16384;

    k_fpad<<<(CCH * 4356 + 255) / 256, 256, 0, stream>>>(fb, fpad);
    k_bpad<<<(CCH * 16900 + 255) / 256, 256, 0, stream>>>(bb, bpad);
    k_gemmS<<<dim3(32, 32), 256, 0, stream>>>(fpad, S);
    k_stats<<<(LTOT + 255) / 256, 256, 0, stream>>>(S, mb, invn, mmv);
    k_softmax<<<LTOT, 256, 0, stream>>>(S, invn, mmv);
    k_gemmR<<<dim3(32, 16), 256, 0, stream>>>(bpad, S, R);
    k_combine<<<(CCH * 16384 + 255) / 256, 256, 0, stream>>>(R, ob);
  }
}